// Loss_29111288332476
// MI455X (gfx1250) — compile-verified
//
#include <hip/hip_runtime.h>

#define NUM_CLASS 1000
#define FEAT_DIM  512
#define BATCH     131072
#define TILES     63            // ceil(1000/16)
#define NBLK      (TILES*TILES) // 3969 partial sums

typedef float v2f __attribute__((ext_vector_type(2)));
typedef float v8f __attribute__((ext_vector_type(8)));

// ---------------------------------------------------------------------------
// n2[i] = sum(X[i,:]^2), one wave per row.
// ---------------------------------------------------------------------------
__global__ __launch_bounds__(32)
void norms_kernel(const float* __restrict__ X, float* __restrict__ n2) {
    const int row  = blockIdx.x;
    const int lane = threadIdx.x;
    const float* r = X + (size_t)row * FEAT_DIM;
    float s = 0.f;
#pragma unroll
    for (int t = 0; t < FEAT_DIM / 32; ++t) {
        float x = r[lane + 32 * t];
        s += x * x;
    }
#pragma unroll
    for (int off = 16; off > 0; off >>= 1)
        s += __shfl_xor(s, off, 32);
    if (lane == 0) n2[row] = s;
}

// ---------------------------------------------------------------------------
// One wave per 16x16 tile of the Gram matrix G = X * X^T via
// V_WMMA_F32_16X16X4_F32 (K marched in steps of 4, 128 WMMAs per tile).
// Fuses d = sqrt(max(n2[i]+n2[j]-2*G, 0)), optional hinge vs thres, and a
// wave-level reduction into one partial sum per block (deterministic).
//
// f32 WMMA lane layout (ISA 7.12.2):
//   A 16x4 : lanes 0-15 hold row M=lane, K=0(.x),1(.y); lanes 16-31 K=2,3
//   B 4x16 : symmetric (column N = lane%16)
//   C/D    : c[v] is element (M = v + 8*(lane/16), N = lane%16)
// Since B = A^T of the column tile, both tiles load with the identical
// per-lane pattern: row base+(lane&15), cols k + 2*(lane>>4) + {0,1}.
// ---------------------------------------------------------------------------
__global__ __launch_bounds__(32)
void gram_dist_kernel(const float* __restrict__ X,
                      const float* __restrict__ n2,
                      float* __restrict__ partials,
                      const float* __restrict__ thres_ptr,
                      int hinge_mode) {
    const int lane = threadIdx.x;
    const int half = lane >> 4;   // 0: K={0,1}, 1: K={2,3}
    const int m    = lane & 15;
    const int rb   = blockIdx.x * 16;
    const int cb   = blockIdx.y * 16;

    int arow = rb + m; if (arow > NUM_CLASS - 1) arow = NUM_CLASS - 1; // clamp pad rows
    int brow = cb + m; if (brow > NUM_CLASS - 1) brow = NUM_CLASS - 1;

    const float* aptr = X + (size_t)arow * FEAT_DIM + 2 * half;
    const float* bptr = X + (size_t)brow * FEAT_DIM + 2 * half;

    v8f c = {};
#pragma unroll 8
    for (int k = 0; k < FEAT_DIM; k += 4) {
        v2f a = *(const v2f*)(aptr + k);   // 8B-aligned global_load_b64
        v2f b = *(const v2f*)(bptr + k);
        c = __builtin_amdgcn_wmma_f32_16x16x4_f32(
                /*neg_a=*/false, a, /*neg_b=*/false, b,
                /*c_mod=*/(short)0, c, /*reuse_a=*/false, /*reuse_b=*/false);
    }

    const float thres = hinge_mode ? thres_ptr[0] : 0.0f;

    float local = 0.f;
#pragma unroll
    for (int v = 0; v < 8; ++v) {
        const int i = rb + v + 8 * half;
        const int j = cb + m;
        if (i < NUM_CLASS && j < NUM_CLASS) {
            float d2 = n2[i] + n2[j] - 2.0f * c[v];
            d2 = d2 > 0.f ? d2 : 0.f;
            float d = __builtin_sqrtf(d2);
            local += hinge_mode ? (d < thres ? thres - d : 0.f) : d;
        }
    }
#pragma unroll
    for (int off = 16; off > 0; off >>= 1)
        local += __shfl_xor(local, off, 32);
    if (lane == 0) partials[blockIdx.y * gridDim.x + blockIdx.x] = local;
}

// ---------------------------------------------------------------------------
// Deterministic single-block reduction of n partials; out[0] = sum * scale.
// ---------------------------------------------------------------------------
__global__ __launch_bounds__(256)
void reduce_scale_kernel(const float* __restrict__ in, int n, float scale,
                         float* __restrict__ out) {
    __shared__ float sm[256];
    float s = 0.f;
    for (int i = threadIdx.x; i < n; i += 256) s += in[i];
    sm[threadIdx.x] = s;
    __syncthreads();
    for (int off = 128; off > 0; off >>= 1) {
        if (threadIdx.x < (unsigned)off) sm[threadIdx.x] += sm[threadIdx.x + off];
        __syncthreads();
    }
    if (threadIdx.x == 0) out[0] = sm[0] * scale;
}

// ---------------------------------------------------------------------------
// Deterministic segment-mean: one block (8 waves, 256 threads) per class.
// Each wave scans a fixed stripe of labels with a wave32 ballot; matching
// feature rows are read once, fully coalesced (float4 across the wave =
// contiguous 512B). Wave partials combined in LDS in a fixed order -> no
// float atomics, bitwise-deterministic. Writes new_center = (center+sum)/
// max(count,1) directly into the output buffer.
// ---------------------------------------------------------------------------
__global__ __launch_bounds__(256)
void segment_mean_kernel(const float* __restrict__ features,
                         const int*   __restrict__ labels,
                         const float* __restrict__ center,
                         float* __restrict__ newc) {
    __shared__ float ssum[8 * FEAT_DIM];   // 16 KB
    __shared__ int   scnt[8];

    const int cls  = blockIdx.x;
    const int tid  = threadIdx.x;
    const int wave = tid >> 5;
    const int lane = tid & 31;

    float4 a0 = {0,0,0,0}, a1 = {0,0,0,0}, a2 = {0,0,0,0}, a3 = {0,0,0,0};
    int cnt = 0;

    for (int i0 = wave * 32; i0 < BATCH; i0 += 256) {
        const int lab = labels[i0 + lane];
        unsigned mask = __builtin_amdgcn_ballot_w32(lab == cls); // uniform
        cnt += __builtin_popcount(mask);
        while (mask) {
            const int b = __builtin_ctz(mask);
            mask &= mask - 1;
            const float4* row =
                (const float4*)(features + (size_t)(i0 + b) * FEAT_DIM);
            float4 r0 = row[lane];        // cols [  0..127]
            float4 r1 = row[32 + lane];   // cols [128..255]
            float4 r2 = row[64 + lane];   // cols [256..383]
            float4 r3 = row[96 + lane];   // cols [384..511]
            a0.x += r0.x; a0.y += r0.y; a0.z += r0.z; a0.w += r0.w;
            a1.x += r1.x; a1.y += r1.y; a1.z += r1.z; a1.w += r1.w;
            a2.x += r2.x; a2.y += r2.y; a2.z += r2.z; a2.w += r2.w;
            a3.x += r3.x; a3.y += r3.y; a3.z += r3.z; a3.w += r3.w;
        }
    }

    float4* sv = (float4*)ssum;
    sv[wave * 128 +      lane] = a0;
    sv[wave * 128 + 32 + lane] = a1;
    sv[wave * 128 + 64 + lane] = a2;
    sv[wave * 128 + 96 + lane] = a3;
    if (lane == 0) scnt[wave] = cnt;
    __syncthreads();

    int count = 0;
#pragma unroll
    for (int w = 0; w < 8; ++w) count += scnt[w];
    const float denom = fmaxf((float)count, 1.0f);

    for (int col = tid; col < FEAT_DIM; col += 256) {
        float s = 0.f;
#pragma unroll
        for (int w = 0; w < 8; ++w) s += ssum[w * FEAT_DIM + col]; // fixed order
        newc[(size_t)cls * FEAT_DIM + col] =
            (center[(size_t)cls * FEAT_DIM + col] + s) / denom;
    }
}

// ---------------------------------------------------------------------------
extern "C" void kernel_launch(void* const* d_in, const int* in_sizes, int n_in,
                              void* d_out, int out_size, void* d_ws, size_t ws_size,
                              hipStream_t stream) {
    (void)in_sizes; (void)n_in; (void)out_size; (void)ws_size;
    const float* features = (const float*)d_in[0];  // [131072,512] f32
    const float* center   = (const float*)d_in[1];  // [1000,512]   f32
    const int*   labels   = (const int*)  d_in[2];  // [131072]     i32

    float* out   = (float*)d_out;
    float* loss  = out;       // out[0]
    float* newc  = out + 1;   // out[1 .. 512000] = new_center (row-major)

    float* wsf      = (float*)d_ws;
    float* n2       = wsf;            // 1024 floats (1000 used)
    float* partials = wsf + 1024;     // 4096 floats (3969 used)
    float* thres    = wsf + 1024 + 4096; // 1 float

    const dim3 gtiles(TILES, TILES);
    const float inv = 1.0f / (float)(NUM_CLASS * NUM_CLASS);

    // thres = 3 * mean(cdist(center, center))
    norms_kernel<<<NUM_CLASS, 32, 0, stream>>>(center, n2);
    gram_dist_kernel<<<gtiles, 32, 0, stream>>>(center, n2, partials, thres, 0);
    reduce_scale_kernel<<<1, 256, 0, stream>>>(partials, NBLK, 3.0f * inv, thres);

    // new_center = (center + segment_sum(features)) / max(count, 1)
    segment_mean_kernel<<<NUM_CLASS, 256, 0, stream>>>(features, labels, center, newc);

    // inter_loss = mean(relu(thres - cdist(new_center, new_center)))
    norms_kernel<<<NUM_CLASS, 32, 0, stream>>>(newc, n2);
    gram_dist_kernel<<<gtiles, 32, 0, stream>>>(newc, n2, partials, thres, 1);
    reduce_scale_kernel<<<1, 256, 0, stream>>>(partials, NBLK, inv, loss);
}